// allEntEncodingPQC_u3loc_69990787055789
// MI455X (gfx1250) — compile-verified
//
#include <hip/hip_runtime.h>

typedef float v2f __attribute__((ext_vector_type(2)));
typedef float v8f __attribute__((ext_vector_type(8)));

#define NT 256

struct C2 { float r, i; };
__device__ __forceinline__ C2 cmul(C2 a, C2 b){ C2 o; o.r = a.r*b.r - a.i*b.i; o.i = a.r*b.i + a.i*b.r; return o; }
__device__ __forceinline__ C2 cadd(C2 a, C2 b){ C2 o; o.r = a.r + b.r; o.i = a.i + b.i; return o; }

__device__ __forceinline__ void m2mul(C2* o, const C2* A, const C2* B){
  o[0] = cadd(cmul(A[0], B[0]), cmul(A[1], B[2]));
  o[1] = cadd(cmul(A[0], B[1]), cmul(A[1], B[3]));
  o[2] = cadd(cmul(A[2], B[0]), cmul(A[3], B[2]));
  o[3] = cadd(cmul(A[2], B[1]), cmul(A[3], B[3]));
}

// fused U = Rz(az) * Ry(ay) * Rx(ax)  (reference applies rx, ry, rz in sequence)
// stored as 8 floats: r00 i00 r01 i01 r10 i10 r11 i11
__device__ __forceinline__ void fuse3(float ax, float ay, float az, float* u){
  float sa, ca, sb, cb, sg, cg;
  sincosf(0.5f * ax, &sa, &ca);
  sincosf(0.5f * ay, &sb, &cb);
  sincosf(0.5f * az, &sg, &cg);
  C2 Rx[4] = {{ca, 0.f}, {0.f, -sa}, {0.f, -sa}, {ca, 0.f}};
  C2 Ry[4] = {{cb, 0.f}, {-sb, 0.f}, {sb, 0.f}, {cb, 0.f}};
  C2 Rz[4] = {{cg, -sg}, {0.f, 0.f}, {0.f, 0.f}, {cg, sg}};
  C2 T[4], U[4];
  m2mul(T, Ry, Rx);
  m2mul(U, Rz, T);
  #pragma unroll
  for (int k = 0; k < 4; ++k){ u[2*k] = U[k].r; u[2*k+1] = U[k].i; }
}

__device__ __forceinline__ void setI(float* u){
  u[0]=1.f; u[1]=0.f; u[2]=0.f; u[3]=0.f; u[4]=0.f; u[5]=0.f; u[6]=1.f; u[7]=0.f;
}
__device__ __forceinline__ void setH(float* u){
  const float s = 0.70710678118654752440f;
  u[0]=s; u[1]=0.f; u[2]=s; u[3]=0.f; u[4]=s; u[5]=0.f; u[6]=-s; u[7]=0.f;
}

__device__ __forceinline__ v8f wmma4(v2f a, v2f b, v8f c){
  // D = A(16x4,f32) * B(4x16,f32) + C(16x16,f32); exact f32 MAC, RNE
  return __builtin_amdgcn_wmma_f32_16x16x4_f32(false, a, false, b, (short)0, c, false, false);
}

// Apply 2x2 complex u to amplitude pair (a0, a1) held in LDS (split re/im).
__device__ __forceinline__ void apair(float* sRe, float* sIm, int a0, int a1, const float* u){
  float xr = sRe[a0], xi = sIm[a0];
  float yr = sRe[a1], yi = sIm[a1];
  sRe[a0] = u[0]*xr - u[1]*xi + u[2]*yr - u[3]*yi;
  sIm[a0] = u[0]*xi + u[1]*xr + u[2]*yi + u[3]*yr;
  sRe[a1] = u[4]*xr - u[5]*xi + u[6]*yr - u[7]*yi;
  sIm[a1] = u[4]*xi + u[5]*xr + u[6]*yi + u[7]*yr;
}

// Build 16x16 group unitary U = g[0] (x) g[1] (x) g[2] (x) g[3] from four 2x2 gates in g4,
// then apply it to the state viewed as 16 x 256 for contiguous qubit group p..p+3,
// using complex f32 WMMA (4 real 16x16x16 matmuls per 16-col tile).
// Callers must have a __syncthreads() between filling g4 and calling this.
__device__ void applyGroup(float* sRe, float* sIm, float* uRe, float* uIm,
                           const float* g4, int p, int tid){
  { // build kron product: pos 0 = MSB of the 4-bit row (qubit p has largest weight)
    int r = tid >> 4, c = tid & 15;
    C2 prod = {1.f, 0.f};
    #pragma unroll
    for (int pos = 0; pos < 4; ++pos){
      int rb = (r >> (3 - pos)) & 1;
      int cb = (c >> (3 - pos)) & 1;
      C2 g = { g4[pos*8 + (rb*2 + cb)*2], g4[pos*8 + (rb*2 + cb)*2 + 1] };
      prod = cmul(prod, g);
    }
    uRe[tid] = prod.r; uIm[tid] = prod.i;
  }
  __syncthreads();

  const int lane = tid & 31;
  const int wave = tid >> 5;
  const int m16  = lane & 15;
  const int hib  = lane >> 4;      // half-wave select
  const int koff = hib * 2;        // A/B K split: lanes<16 -> K 0,1 ; lanes>=16 -> K 2,3
  const int shf  = 8 - p;
  const int W    = 1 << shf;       // weight of the group's LSB qubit

  // A fragments of U (row-major 16x16 in LDS); Ai pre-negated for yr accumulation
  v2f ar[4], ai[4], an[4];
  #pragma unroll
  for (int k = 0; k < 4; ++k){
    int ui = m16*16 + 4*k + koff;
    v2f t0 = { uRe[ui], uRe[ui+1] };  ar[k] = t0;
    v2f t1 = { uIm[ui], uIm[ui+1] };  ai[k] = t1;
    v2f t2 = { -uIm[ui], -uIm[ui+1] }; an[k] = t2;
  }

  // two 16-column tiles per wave; columns are disjoint across waves and tiles
  #pragma unroll
  for (int t = 0; t < 2; ++t){
    int c    = (wave*2 + t)*16 + m16;       // this lane's column (B/D "N" index)
    int hi   = c >> shf;
    int lo   = c & (W - 1);
    int base = (hi << (shf + 4)) + lo;      // addr(r, c) = base + r*W

    v2f br[4], bi[4];
    #pragma unroll
    for (int k = 0; k < 4; ++k){
      int r0 = 4*k + koff;
      v2f t0 = { sRe[base + r0*W], sRe[base + (r0+1)*W] }; br[k] = t0;
      v2f t1 = { sIm[base + r0*W], sIm[base + (r0+1)*W] }; bi[k] = t1;
    }

    v8f aR = {}; v8f aI = {};
    #pragma unroll
    for (int k = 0; k < 4; ++k) aR = wmma4(ar[k], br[k], aR);  // +Ur*xr
    #pragma unroll
    for (int k = 0; k < 4; ++k) aR = wmma4(an[k], bi[k], aR);  // -Ui*xi
    #pragma unroll
    for (int k = 0; k < 4; ++k) aI = wmma4(ar[k], bi[k], aI);  // +Ur*xi
    #pragma unroll
    for (int k = 0; k < 4; ++k) aI = wmma4(ai[k], br[k], aI);  // +Ui*xr

    #pragma unroll
    for (int v = 0; v < 8; ++v){
      int m = v + hib*8;                    // C/D layout: VGPR v -> rows v / v+8
      sRe[base + m*W] = aR[v];
      sIm[base + m*W] = aI[v];
    }
  }
  __syncthreads();
}

// One fused conv_layer: for every amplitude with bit(tq)==1, apply on target rq the
// 2x2 gate selected by (bit yq, bit xq, bit kq=9); all 8 control combos are disjoint.
__device__ void cstep(float* sRe, float* sIm, const float* convU,
                      int tq, int rq, int yq, int xq, int tid){
  int wt = 1 << (11 - tq);
  int wr = 1 << (11 - rq);     // wr < wt for all call sites
  for (int p = tid; p < 1024; p += NT){
    int t1 = ((p  & ~(wr - 1)) << 1) | (p  & (wr - 1));          // insert 0 at rq bit
    int a0 = (((t1 & ~(wt - 1)) << 1) | (t1 & (wt - 1))) | wt;   // insert 1 at tq bit
    int a1 = a0 | wr;
    int ab = (a0 >> (11 - yq)) & 1;
    int bb = (a0 >> (11 - xq)) & 1;
    int cv = (a0 >> 2) & 1;                                      // kq = 9 -> addr bit 2
    apair(sRe, sIm, a0, a1, convU + (cv*4 + ab + 2*bb) * 8);
  }
}

__global__ __launch_bounds__(NT)
void qsim12_kernel(const float* __restrict__ inputs, const float* __restrict__ locp,
                   const float* __restrict__ conv,   float* __restrict__ out){
  __shared__ float sRe[4096];
  __shared__ float sIm[4096];
  __shared__ float uRe[256];
  __shared__ float uIm[256];
  __shared__ float g4[32];        // four 2x2 complex gates for the current group unitary
  __shared__ float encU[192*8];   // encoding: 64 cells x 3 targets fused gates
  __shared__ float convU[8*8];    // conv layer: 2 cv x 4 loc-states fused gates

  const int tid = threadIdx.x;
  const float* IM = inputs + (size_t)blockIdx.x * 576;

  // |0...0>
  for (int i = tid; i < 4096; i += NT){ sRe[i] = 0.f; sIm[i] = 0.f; }
  if (tid == 0) sRe[0] = 1.f;

  // ---- H on qubits 0..3 (group p=0) ----
  if (tid < 4) setH(g4 + tid*8);
  __syncthreads();
  applyGroup(sRe, sIm, uRe, uIm, g4, 0, tid);

  // ---- H on qubits 4,5 (group p=4: H,H,I,I) ----
  if (tid < 4){ if (tid < 2) setH(g4 + tid*8); else setI(g4 + tid*8); }
  __syncthreads();
  applyGroup(sRe, sIm, uRe, uIm, g4, 4, tid);

  // ---- encoding: 64 disjoint 64-amp subspaces (controls = top 6 address bits) ----
  if (tid < 192){
    int cell = tid / 3, n = tid - 3*cell;
    // addr-bit mapping: cell bit5..0 = (j0, j1, j2, i0, i1, i2)
    int i = ((cell >> 2) & 1) | (((cell >> 1) & 1) << 1) | ((cell & 1) << 2);
    int j = ((cell >> 5) & 1) | (((cell >> 4) & 1) << 1) | (((cell >> 3) & 1) << 2);
    int b0 = 9 * (8*i + j) + 3*n;
    fuse3(IM[b0], IM[b0+1], IM[b0+2], encU + tid*8);
  }
  __syncthreads();
  #pragma unroll
  for (int n = 0; n < 3; ++n){                 // targets qubit 6,7,8 (cell-local weights 32,16,8)
    int w = 32 >> n;
    for (int q = tid; q < 2048; q += NT){
      int cell = q >> 5, pl = q & 31;
      int idx0 = ((pl & ~(w - 1)) << 1) | (pl & (w - 1));
      int a0 = cell*64 + idx0;
      apair(sRe, sIm, a0, a0 + w, encU + (cell*3 + n)*8);
    }
    __syncthreads();
  }
  // CZ(6,7), CZ(7,8), CZ(8,6) => sign by pairwise-AND parity of bits 5,4,3
  for (int a = tid; a < 4096; a += NT){
    int b6 = (a >> 5) & 1, b7 = (a >> 4) & 1, b8 = (a >> 3) & 1;
    if ((b6 & b7) ^ (b7 & b8) ^ (b8 & b6)){ sRe[a] = -sRe[a]; sIm[a] = -sIm[a]; }
  }

  // ---- H on kernel qubit 9 (group p=8: I,H,I,I) ----
  if (tid < 4){ if (tid == 1) setH(g4 + 8); else setI(g4 + tid*8); }
  __syncthreads();
  applyGroup(sRe, sIm, uRe, uIm, g4, 8, tid);

  // ---- qdcnn layers ----
  for (int j3 = 0; j3 < 3; ++j3){
    const float* L  = locp + 18*j3;
    const float* CV = conv + 48*j3;

    // loc_conv(qa=3, qb=0): group p=0, pos0 = u_q0 (L[3..5]), pos3 = u_q3 (L[0..2])
    if (tid < 4){
      if (tid == 0)      fuse3(L[3], L[4], L[5], g4);
      else if (tid == 3) fuse3(L[0], L[1], L[2], g4 + 24);
      else               setI(g4 + tid*8);
    }
    __syncthreads();
    applyGroup(sRe, sIm, uRe, uIm, g4, 0, tid);

    // conv_layer(xq=3, yq=0, tq=6+j, kq=9, rq=10)
    if (tid < 8){
      int cv = tid >> 2, i4 = tid & 3;
      int o = 12*cv + 3*i4;
      fuse3(CV[o], CV[o+1], CV[o+2], convU + tid*8);
    }
    __syncthreads();
    cstep(sRe, sIm, convU, 6 + j3, 10, 0, 3, tid);
    __syncthreads();

    // loc_conv(qa=4, qb=1): group p=1, pos0 = u_q1 (L[9..11]), pos3 = u_q4 (L[6..8])
    if (tid < 4){
      if (tid == 0)      fuse3(L[9],  L[10], L[11], g4);
      else if (tid == 3) fuse3(L[6],  L[7],  L[8],  g4 + 24);
      else               setI(g4 + tid*8);
    }
    __syncthreads();
    applyGroup(sRe, sIm, uRe, uIm, g4, 1, tid);

    // conv_layer(xq=4, yq=1, tq=10, kq=9, rq=11)
    if (tid < 8){
      int cv = tid >> 2, i4 = tid & 3;
      int o = 24 + 12*cv + 3*i4;
      fuse3(CV[o], CV[o+1], CV[o+2], convU + tid*8);
    }
    __syncthreads();
    cstep(sRe, sIm, convU, 10, 11, 1, 4, tid);
    __syncthreads();

    // loc_conv(qa=5, qb=2): group p=2, pos0 = u_q2 (L[15..17]), pos3 = u_q5 (L[12..14])
    if (tid < 4){
      if (tid == 0)      fuse3(L[15], L[16], L[17], g4);
      else if (tid == 3) fuse3(L[12], L[13], L[14], g4 + 24);
      else               setI(g4 + tid*8);
    }
    __syncthreads();
    applyGroup(sRe, sIm, uRe, uIm, g4, 2, tid);
  }

  // ---- measurement-basis H on (2,5,6,7,8,9,11) ----
  if (tid < 4){ if (tid == 0 || tid == 3) setH(g4 + tid*8); else setI(g4 + tid*8); }
  __syncthreads();
  applyGroup(sRe, sIm, uRe, uIm, g4, 2, tid);       // H on q2, q5
  if (tid < 4) setH(g4 + tid*8);
  __syncthreads();
  applyGroup(sRe, sIm, uRe, uIm, g4, 6, tid);       // H on q6..q9
  if (tid < 4){ if (tid == 3) setH(g4 + tid*8); else setI(g4 + tid*8); }
  __syncthreads();
  applyGroup(sRe, sIm, uRe, uIm, g4, 8, tid);       // H on q11

  // ---- readout: marginalize over q0,q1,q3,q4,q10; gather 64 observables ----
  if (tid < 64){
    const int IMGB[4][3] = {{0,0,1},{1,0,1},{0,1,1},{1,1,1}};        // (a2,a5,a11)
    const int CHB[8][3]  = {{1,1,1},{0,1,1},{1,0,1},{0,0,1},
                            {1,1,0},{0,1,0},{1,0,0},{0,0,0}};        // (c6,c7,c8)
    int ig = tid >> 4, k9 = (tid >> 3) & 1, ch = tid & 7;
    int base = IMGB[ig][0]*512 + IMGB[ig][1]*64
             + CHB[ch][0]*32 + CHB[ch][1]*16 + CHB[ch][2]*8
             + k9*4 + IMGB[ig][2];
    float s = 0.f;
    #pragma unroll 4
    for (int m = 0; m < 32; ++m){
      int addr = base + ((m >> 4) & 1)*2048 + ((m >> 3) & 1)*1024
                      + ((m >> 2) & 1)*256  + ((m >> 1) & 1)*128 + (m & 1)*2;
      float re = sRe[addr], im = sIm[addr];
      s += re*re + im*im;
    }
    out[(size_t)blockIdx.x * 64 + tid] = 128.f * s;
  }
}

extern "C" void kernel_launch(void* const* d_in, const int* in_sizes, int n_in,
                              void* d_out, int out_size, void* d_ws, size_t ws_size,
                              hipStream_t stream){
  (void)n_in; (void)out_size; (void)d_ws; (void)ws_size;
  const float* inputs = (const float*)d_in[0];   // (32, 576)
  const float* locp   = (const float*)d_in[1];   // (54,)
  const float* conv   = (const float*)d_in[2];   // (144,)
  float* out = (float*)d_out;                    // (32, 64) float32
  int nimg = in_sizes[0] / 576;
  qsim12_kernel<<<nimg, NT, 0, stream>>>(inputs, locp, conv, out);
}